// LuongAttnDecoderRNNInfo_62319975464974
// MI455X (gfx1250) — compile-verified
//
#include <hip/hip_runtime.h>
#include <hip/hip_bf16.h>
#include <math.h>

typedef __attribute__((ext_vector_type(2))) float v2f;
typedef __attribute__((ext_vector_type(8))) float v8f;

#define B_SZ 128
#define S_SZ 4096
#define H_SZ 512
#define V_SZ 32000
#define SC 64                 // encoder rows per attention chunk (128 KB LDS stage)
#define NCHUNK (S_SZ / SC)    // 64

// ---------------------------------------------------------------------------
// CDNA5 async global->LDS copy (ASYNCcnt path, no VGPR landing zone).
// GV mode: global_load_async_to_lds_b128 vdst_lds, vaddr64, off
// ---------------------------------------------------------------------------
__device__ __forceinline__ void async_copy16_to_lds(unsigned lds_byte_addr,
                                                    const void* gaddr) {
    asm volatile("global_load_async_to_lds_b128 %0, %1, off"
                 :: "v"(lds_byte_addr), "v"(gaddr)
                 : "memory");
}

__device__ __forceinline__ void wait_asynccnt0() {
#if defined(__has_builtin) && __has_builtin(__builtin_amdgcn_s_wait_asynccnt)
    __builtin_amdgcn_s_wait_asynccnt(0);
#else
    asm volatile("s_wait_asynccnt 0x0" ::: "memory");
#endif
}

// ---------------------------------------------------------------------------
// Embedding gather: x[b] = [emb_tok[tok[b]] | emb_pos | emb_year | emb_inst]
// ---------------------------------------------------------------------------
__global__ __launch_bounds__(512) void gather_embed_kernel(
    const int* __restrict__ tok, const int* __restrict__ pos,
    const int* __restrict__ yr,  const int* __restrict__ inst,
    const float* __restrict__ etok, const float* __restrict__ epos,
    const float* __restrict__ eyr,  const float* __restrict__ eins,
    float* __restrict__ x)
{
    const int b = blockIdx.x;
    const int h = threadIdx.x;            // 0..511
    float* xb = x + (size_t)b * (4 * H_SZ);
    xb[h]             = etok[(size_t)tok[b]  * H_SZ + h];
    xb[H_SZ + h]      = epos[(size_t)pos[b]  * H_SZ + h];
    xb[2 * H_SZ + h]  = eyr [(size_t)yr[b]   * H_SZ + h];
    xb[3 * H_SZ + h]  = eins[(size_t)inst[b] * H_SZ + h];
}

// ---------------------------------------------------------------------------
// Generic fp32 WMMA GEMM:  C[m,n] = act( sum_k A[m,k] * W[n,k] + bias[n] )
// One wave per 16x16 tile, K-loop of v_wmma_f32_16x16x4_f32.
// ---------------------------------------------------------------------------
__global__ __launch_bounds__(32) void gemm_tn_wmma_kernel(
    const float* __restrict__ A, int lda,
    const float* __restrict__ W, int ldw,
    const float* __restrict__ bias,
    float* __restrict__ C, int ldc,
    int K, int act)
{
    const int mt   = blockIdx.x * 16;
    const int nt   = blockIdx.y * 16;
    const int lane = threadIdx.x;
    const int half = lane >> 4;           // 0 or 1 (K split)
    const int l    = lane & 15;

    const float* arow = A + (size_t)(mt + l) * lda + 2 * half;
    const float* wrow = W + (size_t)(nt + l) * ldw + 2 * half;

    v8f acc = {0.f, 0.f, 0.f, 0.f, 0.f, 0.f, 0.f, 0.f};
    for (int k = 0; k < K; k += 4) {
        if ((k & 63) == 0) {  // uniform lookahead hint -> global_prefetch_b8
            __builtin_prefetch(wrow + k + 64, 0, 1);
            __builtin_prefetch(arow + k + 64, 0, 1);
        }
        v2f a = *(const v2f*)(arow + k);
        v2f w = *(const v2f*)(wrow + k);
        acc = __builtin_amdgcn_wmma_f32_16x16x4_f32(
            /*neg_a=*/false, a, /*neg_b=*/false, w,
            /*c_mod=*/(short)0, acc, /*reuse_a=*/false, /*reuse_b=*/false);
    }

    const float bs = bias ? bias[nt + l] : 0.f;
    #pragma unroll
    for (int g = 0; g < 8; ++g) {
        const int row = mt + g + 8 * half;
        float v = acc[g] + bs;
        if (act == 1) v = tanhf(v);
        C[(size_t)row * ldc + nt + l] = v;
    }
}

// ---------------------------------------------------------------------------
// GRU gates (PyTorch order r,z,n): h_new = (1-z)*n + z*h
// Writes h_new to the output hidden slot AND to cat[:, 0:512].
// ---------------------------------------------------------------------------
__global__ __launch_bounds__(256) void gru_gates_kernel(
    const float* __restrict__ gx, const float* __restrict__ gh,
    const float* __restrict__ h_in,
    float* __restrict__ h_new_out, float* __restrict__ cat)
{
    const int idx = blockIdx.x * 256 + threadIdx.x;   // over B*H
    const int b = idx >> 9;
    const int j = idx & (H_SZ - 1);
    const float* gxb = gx + (size_t)b * 3 * H_SZ;
    const float* ghb = gh + (size_t)b * 3 * H_SZ;
    const float xr = gxb[j],            hr = ghb[j];
    const float xz = gxb[H_SZ + j],     hz = ghb[H_SZ + j];
    const float xn = gxb[2 * H_SZ + j], hn = ghb[2 * H_SZ + j];
    const float r = 1.f / (1.f + __expf(-(xr + hr)));
    const float z = 1.f / (1.f + __expf(-(xz + hz)));
    const float n = tanhf(xn + r * hn);
    const float hnew = (1.f - z) * n + z * h_in[idx];
    h_new_out[idx] = hnew;
    cat[(size_t)b * 2 * H_SZ + j] = hnew;
}

// ---------------------------------------------------------------------------
// Flash-style attention chunk: one workgroup per (chunk, b).
// Stages SC=64 encoder rows (128 KB) in LDS via CDNA5 async global->LDS copies
// (ASYNCcnt), then computes scores + chunk softmax stats + partial context
// from the single LDS-resident copy. One global pass over encoder_outputs.
// ---------------------------------------------------------------------------
__global__ __launch_bounds__(256) void attn_flash_kernel(
    const float* __restrict__ enc, const float* __restrict__ cat,
    float* __restrict__ attn_raw,
    float* __restrict__ ctx_p, float* __restrict__ m_p, float* __restrict__ l_p)
{
    extern __shared__ float lds[];
    float* eL  = lds;                    // SC * H
    float* hv  = lds + SC * H_SZ;        // H
    float* sc  = hv + H_SZ;              // SC
    float* red = sc + SC;                // 2

    const int chunk = blockIdx.x;
    const int b     = blockIdx.y;
    const int tid   = threadIdx.x;

    // Async-stage the encoder chunk: 128 x 16B per row, 2 rows per pass.
    // Direct cache->LDS, no VGPR round-trip; VALU stays free for the
    // co-resident workgroup on this WGP.
    const int rlocal = tid >> 7;         // 0/1
    const int c4     = tid & 127;        // float4 column
    const unsigned eL_base = (unsigned)(size_t)(void*)eL;
    for (int i0 = 0; i0 < SC; i0 += 2) {
        const int i = i0 + rlocal;
        const float* src = enc + ((size_t)(chunk * SC + i) * B_SZ + b) * H_SZ
                               + (size_t)c4 * 4;
        async_copy16_to_lds(eL_base + (unsigned)(i * H_SZ + c4 * 4) * 4u, src);
    }

    for (int i = tid; i < H_SZ; i += 256) hv[i] = cat[(size_t)b * 2 * H_SZ + i];

    wait_asynccnt0();       // this wave's async transfers are in LDS
    __syncthreads();        // ...and every other wave's too

    // Scores: 8 waves x 8 rows; float4 LDS reads (conflict-free), wave32 reduce
    const int wave = tid >> 5, lane = tid & 31;
    const float4* hv4 = (const float4*)hv;
    for (int si = wave; si < SC; si += 8) {
        float accv = 0.f;
        const float4* er4 = (const float4*)(eL + si * H_SZ);
        for (int q = lane; q < H_SZ / 4; q += 32) {
            const float4 e = er4[q], g = hv4[q];
            accv += e.x * g.x + e.y * g.y + e.z * g.z + e.w * g.w;
        }
        for (int off = 16; off > 0; off >>= 1) accv += __shfl_xor(accv, off, 32);
        if (lane == 0) {
            sc[si] = accv;
            attn_raw[(size_t)b * S_SZ + chunk * SC + si] = accv;
        }
    }
    __syncthreads();

    if (tid == 0) {
        float m = -3.0e38f;
        for (int i = 0; i < SC; ++i) m = fmaxf(m, sc[i]);
        float lsum = 0.f;
        for (int i = 0; i < SC; ++i) lsum += __expf(sc[i] - m);
        red[0] = m; red[1] = lsum;
        m_p[b * NCHUNK + chunk] = m;
        l_p[b * NCHUNK + chunk] = lsum;
    }
    __syncthreads();
    const float m = red[0];
    if (tid < SC) sc[tid] = __expf(sc[tid] - m);
    __syncthreads();

    // Partial context from LDS-resident rows (no encoder re-read)
    for (int hh = tid; hh < H_SZ; hh += 256) {
        float a = 0.f;
        for (int s = 0; s < SC; ++s) a += sc[s] * eL[s * H_SZ + hh];
        ctx_p[((size_t)b * NCHUNK + chunk) * H_SZ + hh] = a;
    }
}

// ---------------------------------------------------------------------------
// Combine NCHUNK partials per batch row: global (m,l) + rescaled context
// into cat[:, 512:1024].
// ---------------------------------------------------------------------------
__global__ __launch_bounds__(256) void attn_combine_kernel(
    const float* __restrict__ ctx_p, const float* __restrict__ m_p,
    const float* __restrict__ l_p, float* __restrict__ cat,
    float* __restrict__ m_g, float* __restrict__ l_g)
{
    __shared__ float mp[NCHUNK], lp[NCHUNK], red[2];
    const int b = blockIdx.x, tid = threadIdx.x;
    if (tid < NCHUNK) { mp[tid] = m_p[b * NCHUNK + tid]; lp[tid] = l_p[b * NCHUNK + tid]; }
    __syncthreads();
    if (tid == 0) {
        float m = -3.0e38f;
        for (int c = 0; c < NCHUNK; ++c) m = fmaxf(m, mp[c]);
        float lsum = 0.f;
        for (int c = 0; c < NCHUNK; ++c) lsum += lp[c] * __expf(mp[c] - m);
        red[0] = m; red[1] = lsum;
        m_g[b] = m; l_g[b] = lsum;
    }
    __syncthreads();
    const float m = red[0], invl = 1.f / red[1];
    for (int hh = tid; hh < H_SZ; hh += 256) {
        float a = 0.f;
        for (int c = 0; c < NCHUNK; ++c)
            a += ctx_p[((size_t)b * NCHUNK + c) * H_SZ + hh] * __expf(mp[c] - m);
        cat[(size_t)b * 2 * H_SZ + H_SZ + hh] = a * invl;
    }
}

// ---------------------------------------------------------------------------
// attn_w[b,s] = exp(raw - m_b) / l_b  (in-place on the output attn region)
// ---------------------------------------------------------------------------
__global__ __launch_bounds__(256) void attn_norm_kernel(
    float* __restrict__ attn, const float* __restrict__ m_g,
    const float* __restrict__ l_g)
{
    const int idx = blockIdx.x * 256 + threadIdx.x;   // over B*S
    const int b = idx >> 12;                           // S = 4096
    attn[idx] = __expf(attn[idx] - m_g[b]) / l_g[b];
}

// ---------------------------------------------------------------------------
extern "C" void kernel_launch(void* const* d_in, const int* in_sizes, int n_in,
                              void* d_out, int out_size, void* d_ws, size_t ws_size,
                              hipStream_t stream)
{
    (void)in_sizes; (void)n_in; (void)out_size; (void)ws_size;

    const int*   tok   = (const int*)d_in[0];
    const int*   pos   = (const int*)d_in[1];
    const int*   yr    = (const int*)d_in[2];
    const int*   inst  = (const int*)d_in[3];
    const float* h_in  = (const float*)d_in[4];   // [1,B,H]
    const float* enc   = (const float*)d_in[5];   // [S,B,H]
    const float* etok  = (const float*)d_in[6];
    const float* epos  = (const float*)d_in[7];
    const float* eyr   = (const float*)d_in[8];
    const float* eins  = (const float*)d_in[9];
    const float* W_ih  = (const float*)d_in[10];  // [3H,4H]
    const float* b_ih  = (const float*)d_in[11];
    const float* W_hh  = (const float*)d_in[12];  // [3H,H]
    const float* b_hh  = (const float*)d_in[13];
    const float* W_cc  = (const float*)d_in[14];  // [H,2H]
    const float* b_cc  = (const float*)d_in[15];
    const float* W_out = (const float*)d_in[16];  // [V,H]
    const float* b_out = (const float*)d_in[17];

    float* out   = (float*)d_out;                    // [B,V]
    float* h_new = out + (size_t)B_SZ * V_SZ;        // [1,B,H]
    float* attn  = h_new + (size_t)B_SZ * H_SZ;      // [B,1,S]

    float* ws      = (float*)d_ws;
    float* x       = ws;  ws += (size_t)B_SZ * 4 * H_SZ;
    float* gx      = ws;  ws += (size_t)B_SZ * 3 * H_SZ;
    float* gh      = ws;  ws += (size_t)B_SZ * 3 * H_SZ;
    float* cat     = ws;  ws += (size_t)B_SZ * 2 * H_SZ;   // [h_new | context]
    float* cc_out  = ws;  ws += (size_t)B_SZ * H_SZ;
    float* ctx_p   = ws;  ws += (size_t)B_SZ * NCHUNK * H_SZ;
    float* m_p     = ws;  ws += B_SZ * NCHUNK;
    float* l_p     = ws;  ws += B_SZ * NCHUNK;
    float* m_g     = ws;  ws += B_SZ;
    float* l_g     = ws;  ws += B_SZ;

    // 1) embeddings -> x [B,2048]
    gather_embed_kernel<<<B_SZ, H_SZ, 0, stream>>>(tok, pos, yr, inst,
                                                   etok, epos, eyr, eins, x);
    // 2) GRU input/hidden GEMMs (fp32 WMMA)
    gemm_tn_wmma_kernel<<<dim3(B_SZ / 16, 3 * H_SZ / 16), 32, 0, stream>>>(
        x, 4 * H_SZ, W_ih, 4 * H_SZ, b_ih, gx, 3 * H_SZ, 4 * H_SZ, 0);
    gemm_tn_wmma_kernel<<<dim3(B_SZ / 16, 3 * H_SZ / 16), 32, 0, stream>>>(
        h_in, H_SZ, W_hh, H_SZ, b_hh, gh, 3 * H_SZ, H_SZ, 0);
    // 3) gates -> h_new (output slot + cat low half)
    gru_gates_kernel<<<(B_SZ * H_SZ) / 256, 256, 0, stream>>>(gx, gh, h_in,
                                                              h_new, cat);
    // 4) single-pass flash attention over encoder_outputs (async LDS staging)
    const size_t lds_bytes = (size_t)(SC * H_SZ + H_SZ + SC + 2) * sizeof(float);
    (void)hipFuncSetAttribute((const void*)attn_flash_kernel,
                              hipFuncAttributeMaxDynamicSharedMemorySize,
                              (int)lds_bytes);
    attn_flash_kernel<<<dim3(NCHUNK, B_SZ), 256, lds_bytes, stream>>>(
        enc, cat, attn, ctx_p, m_p, l_p);
    attn_combine_kernel<<<B_SZ, 256, 0, stream>>>(ctx_p, m_p, l_p, cat, m_g, l_g);
    attn_norm_kernel<<<(B_SZ * S_SZ) / 256, 256, 0, stream>>>(attn, m_g, l_g);
    // 5) concat GEMM with fused tanh epilogue
    gemm_tn_wmma_kernel<<<dim3(B_SZ / 16, H_SZ / 16), 32, 0, stream>>>(
        cat, 2 * H_SZ, W_cc, 2 * H_SZ, b_cc, cc_out, H_SZ, 2 * H_SZ, 1);
    // 6) logits GEMM [128,512] x [512,32000] (x-major grid -> W_out L2 reuse)
    gemm_tn_wmma_kernel<<<dim3(B_SZ / 16, V_SZ / 16), 32, 0, stream>>>(
        cc_out, H_SZ, W_out, H_SZ, b_out, out, V_SZ, H_SZ, 0);
}